// GridSample_11879879543068
// MI455X (gfx1250) — compile-verified
//
#include <hip/hip_runtime.h>
#include <hip/hip_bf16.h>

namespace {
constexpr int       kC        = 32;
constexpr int       kInEdge   = 64;
constexpr int       kInHW     = kInEdge * kInEdge;            // 4096
constexpr int       kInDHW    = kInHW * kInEdge;              // 262144
constexpr int       kOutEdge  = 96;
constexpr int       kOutSP    = kOutEdge * kOutEdge * kOutEdge; // 884736
constexpr int       kBatch    = 2;
constexpr long long kTotalSP  = (long long)kBatch * kOutSP;   // 1769472
constexpr int       kTile     = 256;                          // threads per block = 8 wave32
constexpr int       kTilesPB  = 4;                            // tiles per block (double-buffered)
constexpr int       kBlocks   = (int)(kTotalSP / (kTile * kTilesPB)); // 1728, exact
} // namespace

__device__ __forceinline__ int clamp64(int v) {
    v = v < 0 ? 0 : v;
    return v > (kInEdge - 1) ? (kInEdge - 1) : v;
}

// One async 4-byte global -> LDS copy (ASYNCcnt-tracked, CDNA5).
__device__ __forceinline__ void async_g2l_b32(unsigned lds_off, const char* gaddr) {
    asm volatile("global_load_async_to_lds_b32 %0, %1, off"
                 :: "v"(lds_off), "v"(gaddr)
                 : "memory");
}

// Stage one tile's grid coords (12 B/lane) into LDS: 3 async b32 per wave.
__device__ __forceinline__ void stage_tile(const float* __restrict__ grid,
                                           long long sp_base, unsigned lds_base,
                                           int tid) {
    const char* g = (const char*)(grid + (sp_base + tid) * 3);
    unsigned    l = lds_base + (unsigned)tid * 12u;
    async_g2l_b32(l + 0u, g + 0);
    async_g2l_b32(l + 4u, g + 4);
    async_g2l_b32(l + 8u, g + 8);
}

__global__ __launch_bounds__(kTile)
void grid_sample3d_kernel(const float* __restrict__ inp,
                          const float* __restrict__ grid,
                          float* __restrict__ out) {
    __shared__ float sg[2][kTile * 3];

    const int       tid        = threadIdx.x;
    const long long blockStart = (long long)blockIdx.x * (kTile * kTilesPB);

    const unsigned lds0 = (unsigned)(size_t)(&sg[0][0]);
    const unsigned lds1 = (unsigned)(size_t)(&sg[1][0]);

    // Prime the pipeline: tile 0 -> buffer 0. (ASYNCcnt = 3)
    stage_tile(grid, blockStart, lds0, tid);

    for (int t = 0; t < kTilesPB; ++t) {
        if (t + 1 < kTilesPB) {
            // Kick off next tile's grid fetch, then wait only for the current
            // tile's 3 async loads (in-order completion => cnt <= 3 suffices).
            stage_tile(grid, blockStart + (long long)(t + 1) * kTile,
                       ((t + 1) & 1) ? lds1 : lds0, tid);
            asm volatile("s_wait_asynccnt 0x3" ::: "memory");
        } else {
            asm volatile("s_wait_asynccnt 0x0" ::: "memory");
        }

        const float* sgt = sg[t & 1];
        const float  gx  = sgt[tid * 3 + 0];
        const float  gy  = sgt[tid * 3 + 1];
        const float  gz  = sgt[tid * 3 + 2];

        const long long sp = blockStart + (long long)t * kTile + tid;
        const int n = (sp >= (long long)kOutSP) ? 1 : 0;   // kBatch == 2
        const int p = (int)(sp - (long long)n * kOutSP);

        // align_corners=False unnormalize: ((g+1)*64 - 1)/2 = 32*g + 31.5
        const float ix = fmaf(gx, 32.0f, 31.5f);
        const float iy = fmaf(gy, 32.0f, 31.5f);
        const float iz = fmaf(gz, 32.0f, 31.5f);

        const float fx = floorf(ix), fy = floorf(iy), fz = floorf(iz);
        const float tx = ix - fx, ty = iy - fy, tz = iz - fz;
        const int   x0 = (int)fx, y0 = (int)fy, z0 = (int)fz;
        const int   x1 = x0 + 1, y1 = y0 + 1, z1 = z0 + 1;

        // Per-axis weights with zero-padding validity folded in (branch-free).
        const float mx0 = (x0 >= 0 && x0 < kInEdge) ? (1.0f - tx) : 0.0f;
        const float mx1 = (x1 >= 0 && x1 < kInEdge) ? tx          : 0.0f;
        const float my0 = (y0 >= 0 && y0 < kInEdge) ? (1.0f - ty) : 0.0f;
        const float my1 = (y1 >= 0 && y1 < kInEdge) ? ty          : 0.0f;
        const float mz0 = (z0 >= 0 && z0 < kInEdge) ? (1.0f - tz) : 0.0f;
        const float mz1 = (z1 >= 0 && z1 < kInEdge) ? tz          : 0.0f;

        const int cx0 = clamp64(x0), cx1 = clamp64(x1);
        const int cy0 = clamp64(y0), cy1 = clamp64(y1);
        const int cz0 = clamp64(z0), cz1 = clamp64(z1);

        const int bz0 = cz0 * kInHW, bz1 = cz1 * kInHW;
        const int by0 = cy0 * kInEdge, by1 = cy1 * kInEdge;

        // Corner offsets o_{dz}{dy}{dx} and weights
        const int o000 = bz0 + by0 + cx0, o001 = bz0 + by0 + cx1;
        const int o010 = bz0 + by1 + cx0, o011 = bz0 + by1 + cx1;
        const int o100 = bz1 + by0 + cx0, o101 = bz1 + by0 + cx1;
        const int o110 = bz1 + by1 + cx0, o111 = bz1 + by1 + cx1;

        const float zy00 = mz0 * my0, zy01 = mz0 * my1;
        const float zy10 = mz1 * my0, zy11 = mz1 * my1;
        const float w000 = zy00 * mx0, w001 = zy00 * mx1;
        const float w010 = zy01 * mx0, w011 = zy01 * mx1;
        const float w100 = zy10 * mx0, w101 = zy10 * mx1;
        const float w110 = zy11 * mx0, w111 = zy11 * mx1;

        const float* __restrict__ pin  = inp + (size_t)n * kC * kInDHW;
        float*                    pout = out + (size_t)n * kC * kOutSP + p;

        // 32 channels x 8 gathers each; unroll for memory-level parallelism.
        // Input is L2-resident (64 MiB << 192 MiB L2); gathers hit L2.
#pragma unroll 8
        for (int c = 0; c < kC; ++c) {
            const float* __restrict__ pc = pin + (size_t)c * kInDHW;
            float acc;
            acc = w000 * pc[o000];
            acc = fmaf(w001, pc[o001], acc);
            acc = fmaf(w010, pc[o010], acc);
            acc = fmaf(w011, pc[o011], acc);
            acc = fmaf(w100, pc[o100], acc);
            acc = fmaf(w101, pc[o101], acc);
            acc = fmaf(w110, pc[o110], acc);
            acc = fmaf(w111, pc[o111], acc);
            // Write-once 226 MB output: non-temporal so it doesn't evict the
            // L2-resident input that is re-read ~28x.
            __builtin_nontemporal_store(acc, pout + (size_t)c * kOutSP);
        }
    }
}

extern "C" void kernel_launch(void* const* d_in, const int* in_sizes, int n_in,
                              void* d_out, int out_size, void* d_ws, size_t ws_size,
                              hipStream_t stream) {
    (void)in_sizes; (void)n_in; (void)out_size; (void)d_ws; (void)ws_size;
    const float* inp  = (const float*)d_in[0];   // [2,32,64,64,64] f32
    const float* grid = (const float*)d_in[1];   // [2,96,96,96,3]  f32
    float*       out  = (float*)d_out;           // [2,32,96,96,96] f32

    grid_sample3d_kernel<<<kBlocks, kTile, 0, stream>>>(inp, grid, out);
}